// RetinaNet_6734508720732
// MI455X (gfx1250) — compile-verified
//
#include <hip/hip_runtime.h>
#include <math.h>
#include <stdint.h>

// ---------------------------------------------------------------------------
// RetinaNet post-processing: per-class greedy NMS on gfx1250 (MI455X).
//
// One workgroup per class; the class's score column (49104 f32 = 192KB)
// lives in the 320KB WGP LDS (CDNA5-only capacity), staged with the gfx1250
// async global->LDS gather path (global_load_async_to_lds_b32, ASYNCcnt +
// s_wait_asynccnt — confirmed in disasm). K=100 sequential rounds of
// wave32-shuffle argmax + coalesced float4 box reads for IoU suppression.
// Boxes (768KB) stay L2-resident and are warmed with global_prefetch_b8.
// No matmul structure exists in this workload, so WMMA is intentionally not
// used; the gfx1250-specific paths are the async LDS DMA, split wait
// counters, prefetch, and wave32 execution.
//
// Round-3 tuning: fuse the suppression sweep with the next round's argmax
// scan (both are thread-local passes over the same owned LDS chunks), so
// each round does ONE ds_load_b128 sweep instead of two, and 2 barriers
// instead of 3. Halves per-round LDS read traffic (~192KB/round saved).
// ---------------------------------------------------------------------------

#define NANCH      49104     // fixed by the reference ([2, 49104, 80] model)
#define NCLS       80
#define KDET       100
#define SCORE_T    0.05f
#define IOU_T      0.5f
#define TPB        1024      // 32 wave32s per workgroup
#define NPAD       (((NANCH + 4 * TPB - 1) / (4 * TPB)) * 4 * TPB)  // 49152
#define ITEMS4     (NPAD / (4 * TPB))                               // 12
#define NEG_INF    (-__builtin_inff())

#if defined(__HIP_DEVICE_COMPILE__)
#define AS1 __attribute__((address_space(1)))
#define AS3 __attribute__((address_space(3)))
#endif

extern "C" __global__ __launch_bounds__(TPB, 1)
void nms_per_class_kernel(const float* __restrict__ scores,  // [N, C] row-major
                          const float* __restrict__ boxes,   // [N, 4] xyxy
                          float* __restrict__ out)           // [56000] flat
{
    extern __shared__ float s_scores[];   // NPAD floats (196,608 B dynamic LDS)
    float4* s_scores4 = (float4*)s_scores;

    __shared__ float red_v[32];
    __shared__ int   red_i[32];
    __shared__ float s_bx1, s_by1, s_bx2, s_by2, s_ba;
    __shared__ int   s_gi;

    const int tid  = threadIdx.x;
    const int c    = blockIdx.x;           // class id
    const int lane = tid & 31;             // wave32 lane
    const int wave = tid >> 5;

    const float4* __restrict__ boxes4 = (const float4*)boxes;

    // ---- Stage class-c score column into LDS via async gather ------------
    // Tail entries [NANCH, NPAD) are set to -inf so the hot loops below need
    // no bounds checks: the tail can never win argmax and its sub-threshold
    // value short-circuits any box read.
#pragma unroll
    for (int i = 0; i < 4 * ITEMS4; ++i) {
        int idx = i * TPB + tid;
        if (idx < NANCH) {
#if defined(__HIP_DEVICE_COMPILE__) && __has_builtin(__builtin_amdgcn_global_load_async_to_lds_b32)
            __builtin_amdgcn_global_load_async_to_lds_b32(
                (AS1 int*)(uintptr_t)(scores + (size_t)idx * NCLS + c),
                (AS3 int*)(&s_scores[idx]),
                /*offset=*/0, /*cpol=*/0);
#else
            s_scores[idx] = scores[(size_t)idx * NCLS + c];
#endif
            // Warm the (L2-resident, 768KB) box table: global_prefetch_b8.
            __builtin_prefetch(&boxes[(size_t)idx * 4], 0, 1);
        } else {
            s_scores[idx] = NEG_INF;
        }
    }
#if defined(__HIP_DEVICE_COMPILE__)
#if __has_builtin(__builtin_amdgcn_s_wait_asynccnt)
    __builtin_amdgcn_s_wait_asynccnt(0);
#else
    asm volatile("s_wait_asynccnt 0" ::: "memory");
#endif
#endif
    __syncthreads();

    // ---- Output layout (flat tuple, reference return order) --------------
    float*  o_sc = out;                           // [C*K]
    float*  o_cl = out + (size_t)NCLS * KDET;     // [C*K]  class ids (as f32)
    float4* o_bx = (float4*)(out + (size_t)2 * NCLS * KDET); // [C*K, 4]
    float*  o_vd = out + (size_t)6 * NCLS * KDET; // [C*K]  valids (1.0/0.0)

    // ---- Initial per-thread argmax scan (4-wide, ds_load_b128) -----------
    // Ascending-index order + strict '>' preserves jnp.argmax first-max
    // tie-break semantics.
    float bv = NEG_INF;
    int   bi = 0x7fffffff;
#pragma unroll
    for (int i = 0; i < ITEMS4; ++i) {
        int q = i * TPB + tid;
        float4 v4 = s_scores4[q];
        int e = q * 4;
        if (v4.x > SCORE_T && v4.x > bv) { bv = v4.x; bi = e;     }
        if (v4.y > SCORE_T && v4.y > bv) { bv = v4.y; bi = e + 1; }
        if (v4.z > SCORE_T && v4.z > bv) { bv = v4.z; bi = e + 2; }
        if (v4.w > SCORE_T && v4.w > bv) { bv = v4.w; bi = e + 3; }
    }

    // ---- K sequential greedy rounds ---------------------------------------
    for (int k = 0; k < KDET; ++k) {
        // wave32 shuffle argmax reduction of the thread-local candidates
#pragma unroll
        for (int m = 16; m > 0; m >>= 1) {
            float ov = __shfl_xor(bv, m, 32);
            int   oi = __shfl_xor(bi, m, 32);
            if (ov > bv || (ov == bv && oi < bi)) { bv = ov; bi = oi; }
        }
        if (lane == 0) { red_v[wave] = bv; red_i[wave] = bi; }
        __syncthreads();
        if (wave == 0) {
            float rv = red_v[lane];
            int   ri = red_i[lane];
#pragma unroll
            for (int m = 16; m > 0; m >>= 1) {
                float ov = __shfl_xor(rv, m, 32);
                int   oi = __shfl_xor(ri, m, 32);
                if (ov > rv || (ov == rv && oi < ri)) { rv = ov; ri = oi; }
            }
            if (lane == 0) {
                bool keep = (rv > SCORE_T);     // rv == -inf => exhausted
                float4 bb = make_float4(0.f, 0.f, 0.f, 0.f);
                if (keep) bb = boxes4[ri];
                s_gi  = keep ? ri : -1;
                s_bx1 = bb.x; s_by1 = bb.y; s_bx2 = bb.z; s_by2 = bb.w;
                s_ba  = (bb.z - bb.x) * (bb.w - bb.y);

                // Emit this detection slot (zeros when invalid, like the ref:
                // where(valid, s, 0) and boxes[idx]*valid).
                int oidx = c * KDET + k;
                o_sc[oidx] = keep ? rv : 0.0f;
                o_cl[oidx] = (float)c;
                o_vd[oidx] = keep ? 1.0f : 0.0f;
                o_bx[oidx] = bb;
            }
        }
        __syncthreads();

        const int gi = s_gi;
        if (gi < 0) {
            // Class exhausted: every remaining slot is invalid by
            // construction. Zero-fill in parallel and stop.
            for (int kk = k + 1 + tid; kk < KDET; kk += TPB) {
                int oidx = c * KDET + kk;
                o_sc[oidx] = 0.0f;
                o_cl[oidx] = (float)c;
                o_vd[oidx] = 0.0f;
                o_bx[oidx] = make_float4(0.f, 0.f, 0.f, 0.f);
            }
            break;
        }
        if (k + 1 >= KDET) break;   // last slot emitted; no next round

        const float bx1 = s_bx1, by1 = s_by1, bx2 = s_bx2, by2 = s_by2;
        const float ba  = s_ba;

        // ---- FUSED pass: suppress vs winner AND compute next round's ------
        // thread-local argmax in one ds_load_b128 sweep. Both operations are
        // local to this thread's owned chunks, so no extra barrier is needed.
        bv = NEG_INF;
        bi = 0x7fffffff;
#pragma unroll
        for (int i = 0; i < ITEMS4; ++i) {
            int q = i * TPB + tid;
            float4 v4 = s_scores4[q];
            int e = q * 4;
            bool dirty = false;
            float* vv = (float*)&v4;
#pragma unroll
            for (int j = 0; j < 4; ++j) {
                int idx = e + j;
                float v = vv[j];
                if (idx == gi) { vv[j] = NEG_INF; dirty = true; continue; }
                if (v <= SCORE_T) continue;   // dead / sub-threshold / tail
                float4 b = boxes4[idx];
                float ix1 = fmaxf(bx1, b.x);
                float iy1 = fmaxf(by1, b.y);
                float ix2 = fminf(bx2, b.z);
                float iy2 = fminf(by2, b.w);
                float iw  = fmaxf(ix2 - ix1, 0.0f);
                float ih  = fmaxf(iy2 - iy1, 0.0f);
                float inter = iw * ih;
                float area  = (b.z - b.x) * (b.w - b.y);
                float iou   = inter / (ba + area - inter + 1e-8f);
                if (iou > IOU_T) {
                    vv[j] = NEG_INF;
                    dirty = true;
                } else if (v > bv) {          // survivor: next-round candidate
                    bv = v;
                    bi = idx;
                }
            }
            if (dirty) s_scores4[q] = v4;
        }
        // (next iteration's reduction barrier orders red_v/red_i reuse)
    }
}

extern "C" void kernel_launch(void* const* d_in, const int* in_sizes, int n_in,
                              void* d_out, int out_size, void* d_ws, size_t ws_size,
                              hipStream_t stream) {
    (void)in_sizes; (void)n_in; (void)out_size; (void)d_ws; (void)ws_size;
    const float* scores = (const float*)d_in[0];   // [49104, 80] f32
    const float* boxes  = (const float*)d_in[1];   // [49104, 4]  f32
    float* out = (float*)d_out;                    // 56000 f32

    const size_t lds_bytes = (size_t)NPAD * sizeof(float);   // 196,608 B

    // CDNA5 WGP has 320KB LDS; lift the dynamic-LDS cap above the default.
    // (Host-side attribute set; not a stream op, graph-capture safe.)
    (void)hipFuncSetAttribute((const void*)nms_per_class_kernel,
                              hipFuncAttributeMaxDynamicSharedMemorySize,
                              (int)lds_bytes);

    nms_per_class_kernel<<<NCLS, TPB, lds_bytes, stream>>>(scores, boxes, out);
}